// RNADistanceHead_8727373545928
// MI455X (gfx1250) — compile-verified
//
#include <hip/hip_runtime.h>
#include <hip/hip_bf16.h>
#include <math.h>

// RNADistanceHead for MI455X (gfx1250).
// Inputs (setup_inputs order): h[2*384*256] f32, mask[2*384] bool(u8),
// ln_g[514], ln_b[514], W1[514*64], b1[64], W2[64*40], b2[40]  (all f32)
// Output: logits [2,384,384,40] f32.
//
// Algebraic decomposition:
//   x = LN(feats)*g + b;  x@W1 = rstd*(F - mu*GW1) + bW1
//   F = A_i + B_j + sin*R0 + cos*R1, where
//   A_i = (h_i * g[0:256]) @ W1[0:256],  B_j = (h_j * g[256:512]) @ W1[256:512]
//   GW1 = sum_p g[p]*W1[p,:],  bW1 = b1 + sum_p b[p]*W1[p,:]
//   R0 = g[512]*W1[512,:],     R1 = g[513]*W1[513,:]
// Then y = gelu(hidden), logits = y @ W2 + b2  -> done with V_WMMA_F32_16X16X4_F32.
//
// Workspace layout (floats) — requires ws_size >= 103232*4 bytes (~413 KB):
//   [0)      GW1[64]
//   [64)     bW1[64]
//   [128)    R0[64]
//   [192)    R1[64]
//   [256)    b2p[48]          (b2 padded to 48)
//   [320)    W2T[48*64]       (W2 transposed, N padded to 48)
//   [3392)   sA[768*64]       per-position A projections
//   [52544)  sB[768*64]       per-position B projections
//   [101696) sums[768*2]      per-position (sum, sum_sq) of h
// total 103232 floats.

#define LSEQ   384
#define DMODEL 256
#define NBINS  40
#define HID    64
#define NTILE  24              // LSEQ/16
#define NPAIRT 300             // NTILE*(NTILE+1)/2
#define WS_GW1   0
#define WS_BW1   64
#define WS_R0    128
#define WS_R1    192
#define WS_B2P   256
#define WS_W2T   320
#define WS_A     3392
#define WS_B     52544
#define WS_SUM   101696

typedef float v2f __attribute__((ext_vector_type(2)));
typedef float v8f __attribute__((ext_vector_type(8)));

__device__ __forceinline__ float gelu_exact(float x) {
    return 0.5f * x * (1.0f + erff(x * 0.70710678118654752f));
}

// ---------------- Kernel A: fold LN params into MLP constants ----------------
__global__ __launch_bounds__(256) void const_prep_kernel(
    const float* __restrict__ ln_g, const float* __restrict__ ln_b,
    const float* __restrict__ W1,   const float* __restrict__ b1,
    const float* __restrict__ W2,   const float* __restrict__ b2,
    float* __restrict__ ws)
{
    int t = threadIdx.x;
    if (t < HID) {
        float gw = 0.0f, bw = b1[t];
        for (int p = 0; p < 514; ++p) {
            float w = W1[p * HID + t];
            gw += ln_g[p] * w;
            bw += ln_b[p] * w;
        }
        ws[WS_GW1 + t] = gw;
        ws[WS_BW1 + t] = bw;
        ws[WS_R0 + t]  = ln_g[512] * W1[512 * HID + t];
        ws[WS_R1 + t]  = ln_g[513] * W1[513 * HID + t];
    }
    for (int idx = t; idx < 48; idx += 256)
        ws[WS_B2P + idx] = (idx < NBINS) ? b2[idx] : 0.0f;
    for (int idx = t; idx < 48 * HID; idx += 256) {
        int n = idx >> 6, k = idx & 63;
        ws[WS_W2T + idx] = (n < NBINS) ? W2[k * NBINS + n] : 0.0f;
    }
}

// ------------- Kernel B: per-position projections + LN statistics -----------
__global__ __launch_bounds__(64) void pos_proj_kernel(
    const float* __restrict__ h, const float* __restrict__ ln_g,
    const float* __restrict__ W1, float* __restrict__ ws)
{
    __shared__ float hg1[DMODEL], hg2[DMODEL], redS[64], redSS[64];
    int p = blockIdx.x;          // 0 .. B*L-1
    int t = threadIdx.x;
    const float* hp = h + (size_t)p * DMODEL;
    float s = 0.0f, ss = 0.0f;
    for (int d = t; d < DMODEL; d += 64) {
        float v = hp[d];
        s += v; ss += v * v;
        hg1[d] = v * ln_g[d];
        hg2[d] = v * ln_g[DMODEL + d];
    }
    redS[t] = s; redSS[t] = ss;
    __syncthreads();
    float aA = 0.0f, aB = 0.0f;
    for (int d = 0; d < DMODEL; ++d) {
        aA += hg1[d] * W1[d * HID + t];
        aB += hg2[d] * W1[(DMODEL + d) * HID + t];
    }
    ws[WS_A + (size_t)p * HID + t] = aA;
    ws[WS_B + (size_t)p * HID + t] = aB;
    if (t == 0) {
        float S = 0.0f, SS = 0.0f;
        for (int i = 0; i < 64; ++i) { S += redS[i]; SS += redSS[i]; }
        ws[WS_SUM + p * 2 + 0] = S;
        ws[WS_SUM + p * 2 + 1] = SS;
    }
}

// ------------------------- per-16-pair group compute ------------------------
// Pairs (xpos, ypos=ybase+m), m = lane&15. Result: acc[3] v8f in D layout.
__device__ __forceinline__ void compute_group(
    const float* __restrict__ Arow,   // LDS: A projection of x   [64]
    const float* __restrict__ Brow,   // LDS: B projection of y_m [64]
    float s_x, float ss_x, float s_y, float ss_y,
    int xpos, int ypos,
    const float* __restrict__ sC,     // LDS consts: GW1|bW1|R0|R1|b2p
    const float* __restrict__ gW2T,   // global: W2T [48][64]
    int lane, v8f acc[3])
{
    const float inv514 = 1.0f / 514.0f;
    int n0 = lane & 15;
    int h2 = lane >> 4;

    float rel = fabsf((float)(xpos - ypos)) * (1.0f / (float)(LSEQ - 1));
    float sp = sinf(3.14159265358979323846f * rel);
    float cp = cosf(3.14159265358979323846f * rel);
    float mu = (s_x + s_y + sp + cp) * inv514;
    float e2 = (ss_x + ss_y + sp * sp + cp * cp) * inv514;
    float rstd = rsqrtf(e2 - mu * mu + 1e-5f);

#pragma unroll
    for (int t = 0; t < 3; ++t) {
        float bv = sC[WS_B2P + n0 + 16 * t];
#pragma unroll
        for (int r = 0; r < 8; ++r) acc[t][r] = bv;
    }

    v2f af[16];
#pragma unroll
    for (int c = 0; c < 16; ++c) {
        int k = 4 * c + 2 * h2;
        float f0 = Arow[k]     + Brow[k]     + sp * sC[WS_R0 + k]     + cp * sC[WS_R1 + k]     - mu * sC[WS_GW1 + k];
        float f1 = Arow[k + 1] + Brow[k + 1] + sp * sC[WS_R0 + k + 1] + cp * sC[WS_R1 + k + 1] - mu * sC[WS_GW1 + k + 1];
        float x0 = rstd * f0 + sC[WS_BW1 + k];
        float x1 = rstd * f1 + sC[WS_BW1 + k + 1];
        af[c].x = gelu_exact(x0);
        af[c].y = gelu_exact(x1);
    }

#pragma unroll
    for (int c = 0; c < 16; ++c) {
        int k = 4 * c + 2 * h2;
#pragma unroll
        for (int t = 0; t < 3; ++t) {
            v2f bf = *(const v2f*)(gW2T + (size_t)(n0 + 16 * t) * HID + k);
            acc[t] = __builtin_amdgcn_wmma_f32_16x16x4_f32(
                false, af[c], false, bf, (short)0, acc[t], false, false);
        }
    }
}

// --------------- Kernel C: pairwise logits, symmetrized, masked -------------
__global__ __launch_bounds__(256) void pair_head_kernel(
    const unsigned char* __restrict__ mask,
    const float* __restrict__ ws,
    float* __restrict__ out)
{
    __shared__ float sAI[16 * HID], sAJ[16 * HID], sBI[16 * HID], sBJ[16 * HID];
    __shared__ float sSum[64];              // [0..31]: I-tile (s,ss)x16; [32..63]: J-tile
    __shared__ float sC[304];               // GW1|bW1|R0|R1|b2p
    __shared__ float sP[16 * 16 * NBINS];   // forward pre-logits P(i0+il, j0+jl)

    int tid = threadIdx.x;
    int bid = blockIdx.x;
    int b = bid / NPAIRT;
    int r = bid % NPAIRT;
    int ti = 0, rowlen = NTILE;
    while (r >= rowlen) { r -= rowlen; ++ti; --rowlen; }
    int tj = ti + r;
    int i0 = ti * 16, j0 = tj * 16;

    const float* wsA = ws + WS_A;
    const float* wsB = ws + WS_B;
    const float* wsS = ws + WS_SUM;
    const float* gW2T = ws + WS_W2T;

    for (int idx = tid; idx < 16 * HID; idx += 256) {
        int il = idx >> 6, k = idx & 63;
        sAI[idx] = wsA[(size_t)(b * LSEQ + i0 + il) * HID + k];
        sAJ[idx] = wsA[(size_t)(b * LSEQ + j0 + il) * HID + k];
        sBI[idx] = wsB[(size_t)(b * LSEQ + i0 + il) * HID + k];
        sBJ[idx] = wsB[(size_t)(b * LSEQ + j0 + il) * HID + k];
    }
    if (tid < 32) {
        int il = tid & 15, which = tid >> 4;
        int pos = which ? (j0 + il) : (i0 + il);
        sSum[which * 32 + il * 2 + 0] = wsS[(b * LSEQ + pos) * 2 + 0];
        sSum[which * 32 + il * 2 + 1] = wsS[(b * LSEQ + pos) * 2 + 1];
    }
    for (int idx = tid; idx < 304; idx += 256) sC[idx] = ws[idx];
    __syncthreads();

    int wave = tid >> 5, lane = tid & 31;
    int m = lane & 15, h2 = lane >> 4;

    // Forward: P(i0+iLoc, j0+m) for iLoc handled by this wave
    for (int gg = 0; gg < 2; ++gg) {
        int iLoc = wave * 2 + gg;
        v8f acc[3];
        compute_group(&sAI[iLoc * HID], &sBJ[m * HID],
                      sSum[iLoc * 2], sSum[iLoc * 2 + 1],
                      sSum[32 + m * 2], sSum[32 + m * 2 + 1],
                      i0 + iLoc, j0 + m, sC, gW2T, lane, acc);
#pragma unroll
        for (int t = 0; t < 3; ++t) {
            int n = (lane & 15) + 16 * t;
            if (n < NBINS) {
#pragma unroll
                for (int rr = 0; rr < 8; ++rr) {
                    int mr = rr + 8 * h2;                 // j-local index
                    sP[(iLoc * 16 + mr) * NBINS + n] = acc[t][rr];
                }
            }
        }
    }
    __syncthreads();

    // Reverse: Q(j0+jLoc, i0+m); combine with P and write both outputs.
    for (int gg = 0; gg < 2; ++gg) {
        int jLoc = wave * 2 + gg;
        v8f acc[3];
        compute_group(&sAJ[jLoc * HID], &sBI[m * HID],
                      sSum[32 + jLoc * 2], sSum[32 + jLoc * 2 + 1],
                      sSum[m * 2], sSum[m * 2 + 1],
                      j0 + jLoc, i0 + m, sC, gW2T, lane, acc);
        int jg = j0 + jLoc;
        unsigned char mj = mask[b * LSEQ + jg];
#pragma unroll
        for (int t = 0; t < 3; ++t) {
            int n = (lane & 15) + 16 * t;
            if (n < NBINS) {
#pragma unroll
                for (int rr = 0; rr < 8; ++rr) {
                    int im = rr + 8 * h2;                 // i-local index
                    int ig = i0 + im;
                    float v = 0.5f * (sP[(im * 16 + jLoc) * NBINS + n] + acc[t][rr]);
                    bool pm = mask[b * LSEQ + ig] && mj;
                    v = pm ? v : 0.0f;
                    out[(((size_t)b * LSEQ + ig) * LSEQ + jg) * NBINS + n] = v;
                    out[(((size_t)b * LSEQ + jg) * LSEQ + ig) * NBINS + n] = v;
                }
            }
        }
    }
}

extern "C" void kernel_launch(void* const* d_in, const int* in_sizes, int n_in,
                              void* d_out, int out_size, void* d_ws, size_t ws_size,
                              hipStream_t stream) {
    const float*         h    = (const float*)d_in[0];
    const unsigned char* mask = (const unsigned char*)d_in[1];
    const float*         ln_g = (const float*)d_in[2];
    const float*         ln_b = (const float*)d_in[3];
    const float*         W1   = (const float*)d_in[4];
    const float*         b1   = (const float*)d_in[5];
    const float*         W2   = (const float*)d_in[6];
    const float*         b2   = (const float*)d_in[7];
    float* ws  = (float*)d_ws;     // needs >= 103232 floats (~413 KB)
    float* out = (float*)d_out;

    const_prep_kernel<<<1, 256, 0, stream>>>(ln_g, ln_b, W1, b1, W2, b2, ws);
    pos_proj_kernel<<<2 * LSEQ, 64, 0, stream>>>(h, ln_g, W1, ws);
    pair_head_kernel<<<2 * NPAIRT, 256, 0, stream>>>(mask, ws, out);
}